// GeoChebyConv_54451595379153
// MI455X (gfx1250) — compile-verified
//
#include <hip/hip_runtime.h>

#define N_NODES 268
#define N_PAD   272
#define E_EDGES 8576
#define F_INC   268
#define H_DIM   16
#define C_DIM   10
#define NTHREADS 256

typedef __attribute__((ext_vector_type(2))) float v2f;
typedef __attribute__((ext_vector_type(8))) float v8f;

// LDS layout (float offsets). Regions reused across phases (barriers between).
//  [0,272)        : dis (degree -> rsqrt in place)
//  [272,13136)    : phase1: Wcat [268][48]
//  [272,4624)     : phase2: t1 [272][16]     (Wcat dead)
//  [4624,8976)    : phase2: t2 [272][16]
//  [13136,26192)  : U = x@Wcat [272][48]
//  [26192,30544)  : h [272][16]
//  [272,8976)     : phase3: V [272][32]      (t1/t2 dead)
//  [8976,11696)   : s1 [272][10]
//  [11696,14416)  : s2 [272][10]             (U dead by now)
//  [14416,17136)  : o  [272][10]
//  [17136,17636)  : r  [10][50]
//  [30544,31056)  : Wbcat [16][32] (zero-padded cols 30..31) -- lives phase0..phase3
#define SMEM_FLOATS 31056

__device__ __forceinline__ void lmul_acc(const int* __restrict__ rowI,
                                         const int* __restrict__ colI,
                                         const float* __restrict__ ewg,
                                         const float* sDis,
                                         const float* src, int sstride,
                                         float* dst, int dstride,
                                         int F, float scale, int tid)
{
    // 16 lanes cooperate per edge; edge-major so dst atomics hit consecutive banks.
    for (int e0 = 0; e0 < E_EDGES; e0 += NTHREADS / 16) {
        int e = e0 + (tid >> 4);
        int f = tid & 15;
        int r = rowI[e];
        int c = colI[e];
        float nrm = -ewg[e] * sDis[r] * sDis[c];
        if (f < F)
            atomicAdd(&dst[r * dstride + f], scale * nrm * src[c * sstride + f]);
    }
}

extern "C" __global__ __launch_bounds__(NTHREADS)
void cheb_graph_kernel(const float* __restrict__ x,
                       const int*   __restrict__ eidx,
                       const float* __restrict__ ew,
                       const float* __restrict__ gc1w,
                       const float* __restrict__ gc1b,
                       const float* __restrict__ gc4w,
                       const float* __restrict__ gc4b,
                       const float* __restrict__ w1,
                       const float* __restrict__ b1,
                       const float* __restrict__ w2,
                       const float* __restrict__ b2,
                       float* __restrict__ out)
{
    extern __shared__ float smem[];
    const int g    = blockIdx.x;
    const int tid  = threadIdx.x;
    const int lane = tid & 31;
    const int wave = tid >> 5;
    const int half = lane >> 4;   // 0: lanes 0-15, 1: lanes 16-31
    const int l16  = lane & 15;

    const float* xg   = x  + (size_t)g * N_NODES * F_INC;
    const int*   rowI = eidx + (size_t)g * 2 * E_EDGES;
    const int*   colI = rowI + E_EDGES;
    const float* ewg  = ew + (size_t)g * E_EDGES;

    float* sDis = smem;
    float* sW   = smem + 272;
    float* sT1  = smem + 272;
    float* sT2  = smem + 4624;
    float* sU   = smem + 13136;
    float* sH   = smem + 26192;
    float* sV   = smem + 272;
    float* sS1  = smem + 8976;
    float* sS2  = smem + 11696;
    float* sO   = smem + 14416;
    float* sR   = smem + 17136;
    float* sWb  = smem + 30544;

    // ---------------- Phase 0: degree -> dis = rsqrt(deg); stage weights ----------------
    for (int i = tid; i < N_PAD; i += NTHREADS) sDis[i] = 0.0f;
    __syncthreads();
    for (int e = tid; e < E_EDGES; e += NTHREADS)
        atomicAdd(&sDis[rowI[e]], ewg[e]);
    __syncthreads();
    for (int i = tid; i < N_PAD; i += NTHREADS) {
        float d = sDis[i];
        sDis[i] = (d > 0.0f) ? rsqrtf(d) : 0.0f;
    }
    // Wcat[268][48]: col block kk holds gc1_w[kk] (16 cols each)
    for (int i = tid; i < F_INC * 48; i += NTHREADS) {
        int k  = i / 48;
        int c  = i - k * 48;
        int kk = c >> 4;
        int hh = c & 15;
        sW[i] = gc1w[(kk * F_INC + k) * H_DIM + hh];
    }
    // Wbcat[16][32]: cols 0..29 = gc4_w[kk][k][cc], cols 30..31 = 0
    for (int i = tid; i < 16 * 32; i += NTHREADS) {
        int k = i >> 5;
        int n = i & 31;
        float v = 0.0f;
        if (n < 30) {
            int kk = n / 10;
            int cc = n - kk * 10;
            v = gc4w[(kk * H_DIM + k) * C_DIM + cc];
        }
        sWb[i] = v;
    }
    __syncthreads();

    // ---------------- Phase 1: U = x @ Wcat  (fp32 WMMA 16x16x4) ----------------
    // Each wave owns an M-strip of 16 rows and computes all 3 N-tiles:
    // one A-fragment load feeds 3 WMMAs; row guard folded into a 0/1 mask.
    for (int mt = wave; mt < 17; mt += 8) {
        int m = mt * 16 + l16;
        const float rmask = (m < N_NODES) ? 1.0f : 0.0f;
        const float* arow = xg + (size_t)((m < N_NODES) ? m : (N_NODES - 1)) * F_INC;
        v8f acc0 = {}, acc1 = {}, acc2 = {};
        #pragma unroll 2
        for (int k0 = 0; k0 < F_INC; k0 += 4) {
            int koff = k0 + 2 * half;   // lanes 0-15: K=k0,k0+1; 16-31: K=k0+2,k0+3
            v2f a = *(const v2f*)(arow + koff);
            a.x *= rmask; a.y *= rmask;
            const float* w0 = sW + koff * 48;
            const float* w1p = sW + (koff + 1) * 48;
            v2f b0, b1c, b2c;
            b0.x  = w0[l16];       b0.y  = w1p[l16];
            b1c.x = w0[16 + l16];  b1c.y = w1p[16 + l16];
            b2c.x = w0[32 + l16];  b2c.y = w1p[32 + l16];
            acc0 = __builtin_amdgcn_wmma_f32_16x16x4_f32(
                false, a, false, b0,  (short)0, acc0, false, false);
            acc1 = __builtin_amdgcn_wmma_f32_16x16x4_f32(
                false, a, false, b1c, (short)0, acc1, false, false);
            acc2 = __builtin_amdgcn_wmma_f32_16x16x4_f32(
                false, a, false, b2c, (short)0, acc2, false, false);
        }
        int rbase = (mt * 16 + 8 * half) * 48 + l16;
        #pragma unroll
        for (int i = 0; i < 8; ++i) {
            sU[rbase + i * 48]      = acc0[i];
            sU[rbase + i * 48 + 16] = acc1[i];
            sU[rbase + i * 48 + 32] = acc2[i];
        }
    }
    __syncthreads();

    // ---------------- Phase 2: h = relu(u0 + L u1 + 2 L(L u2) - u2 + b) ----------------
    for (int i = tid; i < N_PAD * H_DIM; i += NTHREADS) { sT1[i] = 0.0f; sT2[i] = 0.0f; }
    __syncthreads();
    lmul_acc(rowI, colI, ewg, sDis, sU + 16, 48, sT1, 16, 16, 1.0f, tid);  // t1 = L u1
    lmul_acc(rowI, colI, ewg, sDis, sU + 32, 48, sT2, 16, 16, 1.0f, tid);  // t2 = L u2
    __syncthreads();
    for (int i = tid; i < N_PAD * H_DIM; i += NTHREADS) {
        int r = i >> 4, f = i & 15;
        float v = 0.0f;
        if (r < N_NODES)
            v = sU[r * 48 + f] + sT1[i] - sU[r * 48 + 32 + f] + gc1b[f];
        sH[i] = v;
    }
    __syncthreads();
    lmul_acc(rowI, colI, ewg, sDis, sT2, 16, sH, 16, 16, 2.0f, tid);       // h += 2 L t2
    __syncthreads();
    for (int i = tid; i < N_PAD * H_DIM; i += NTHREADS)
        sH[i] = fmaxf(sH[i], 0.0f);
    __syncthreads();

    // ---------------- Phase 3: V = h @ Wbcat (guard-free: sH pad rows are 0) ----------------
    for (int mt = wave; mt < 17; mt += 8) {
        int m = mt * 16 + l16;
        v8f acc0 = {}, acc1 = {};
        #pragma unroll
        for (int k0 = 0; k0 < H_DIM; k0 += 4) {
            int koff = k0 + 2 * half;
            v2f a = *(const v2f*)(sH + m * 16 + koff);
            const float* w0 = sWb + koff * 32;
            const float* w1p = sWb + (koff + 1) * 32;
            v2f b0, b1c;
            b0.x  = w0[l16];      b0.y  = w1p[l16];
            b1c.x = w0[16 + l16]; b1c.y = w1p[16 + l16];
            acc0 = __builtin_amdgcn_wmma_f32_16x16x4_f32(
                false, a, false, b0,  (short)0, acc0, false, false);
            acc1 = __builtin_amdgcn_wmma_f32_16x16x4_f32(
                false, a, false, b1c, (short)0, acc1, false, false);
        }
        int rbase = (mt * 16 + 8 * half) * 32 + l16;
        #pragma unroll
        for (int i = 0; i < 8; ++i) {
            sV[rbase + i * 32]      = acc0[i];
            sV[rbase + i * 32 + 16] = acc1[i];
        }
    }
    __syncthreads();

    // ---------------- Phase 3b: o = v0 + L v1 + 2 L(L v2) - v2 + b ----------------
    for (int i = tid; i < N_PAD * C_DIM; i += NTHREADS) { sS1[i] = 0.0f; sS2[i] = 0.0f; }
    __syncthreads();
    lmul_acc(rowI, colI, ewg, sDis, sV + 10, 32, sS1, 10, 10, 1.0f, tid);  // s1 = L v1
    lmul_acc(rowI, colI, ewg, sDis, sV + 20, 32, sS2, 10, 10, 1.0f, tid);  // s2 = L v2
    __syncthreads();
    for (int i = tid; i < N_NODES * C_DIM; i += NTHREADS) {
        int r = i / 10, f = i - r * 10;
        sO[i] = sV[r * 32 + f] + sS1[i] - sV[r * 32 + 20 + f] + gc4b[f];
    }
    __syncthreads();
    lmul_acc(rowI, colI, ewg, sDis, sS2, 10, sO, 10, 10, 2.0f, tid);       // o += 2 L s2
    __syncthreads();

    // ---------------- Phase 4: classifier d = relu(o^T @ W1 + b1) @ W2 + b2 ----------------
    for (int t = tid; t < C_DIM * 50; t += NTHREADS) {
        int m = t / 50, j = t - m * 50;
        float acc = b1[j];
        for (int k = 0; k < N_NODES; ++k)
            acc += sO[k * C_DIM + m] * w1[k * 50 + j];
        sR[t] = fmaxf(acc, 0.0f);
    }
    __syncthreads();
    if (tid < C_DIM) {
        float acc = b2[0];
        #pragma unroll 10
        for (int j = 0; j < 50; ++j)
            acc += sR[tid * 50 + j] * w2[j];
        out[(size_t)g * C_DIM + tid] = acc;
    }
}

extern "C" void kernel_launch(void* const* d_in, const int* in_sizes, int n_in,
                              void* d_out, int out_size, void* d_ws, size_t ws_size,
                              hipStream_t stream) {
    const float* x   = (const float*)d_in[0];
    const int*   ei  = (const int*)  d_in[1];
    const float* ea  = (const float*)d_in[2];
    const float* g1w = (const float*)d_in[3];
    const float* g1b = (const float*)d_in[4];
    const float* g4w = (const float*)d_in[5];
    const float* g4b = (const float*)d_in[6];
    const float* cw1 = (const float*)d_in[7];
    const float* cb1 = (const float*)d_in[8];
    const float* cw2 = (const float*)d_in[9];
    const float* cb2 = (const float*)d_in[10];
    float* out = (float*)d_out;

    const size_t shmem = SMEM_FLOATS * sizeof(float);
    cheb_graph_kernel<<<dim3(1024), dim3(NTHREADS), shmem, stream>>>(
        x, ei, ea, g1w, g1b, g4w, g4b, cw1, cb1, cw2, cb2, out);
}